// VectorQuantizer_77601469104664
// MI455X (gfx1250) — compile-verified
//
#include <hip/hip_runtime.h>
#include <hip/hip_bf16.h>

typedef float v2f __attribute__((ext_vector_type(2)));
typedef float v8f __attribute__((ext_vector_type(8)));

#define N_ROWS   65536   // 64 * 32 * 32
#define EMB_K    1024
#define EMB_D    64
#define HW       1024    // 32*32
#define N_ELEMS  4194304 // 64*64*32*32

// ---------------------------------------------------------------------------
// Zero the histogram counts.
__global__ void vq_init(unsigned* __restrict__ counts) {
    int t = blockIdx.x * blockDim.x + threadIdx.x;
    if (t < EMB_K) counts[t] = 0u;
}

// ---------------------------------------------------------------------------
// halfnorm[k] = 0.5 * ||e_k||^2
__global__ void vq_norms(const float* __restrict__ emb, float* __restrict__ halfnorm) {
    int k = blockIdx.x * blockDim.x + threadIdx.x;
    if (k >= EMB_K) return;
    const float4* e = (const float4*)(emb + k * EMB_D);
    float s = 0.f;
#pragma unroll
    for (int i = 0; i < EMB_D / 4; ++i) {
        float4 v = e[i];
        s += v.x * v.x + v.y * v.y + v.z * v.z + v.w * v.w;
    }
    halfnorm[k] = 0.5f * s;
}

// ---------------------------------------------------------------------------
// WMMA argmin kernel. One wave handles TWO 16-row M-tiles (32 rows), looping
// over all 64 code tiles of 16 codes. Each B fragment feeds two WMMAs
// (2 WMMA : 1 global_load_b64) and the two accumulator chains are independent.
// score = x.e - 0.5*||e||^2 ; argmin dist == argmax score. The -0.5||e||^2
// bias is preloaded into the accumulator (every slot of a lane shares one
// column), so no post-loop subtraction is needed.
//
// A (16x4 f32) per ISA layout: lane l -> m = l&15, kh = l>>4,
//   vgpr j holds A[m][kbase + 2*kh + j].
// B (4x16 f32): lane l -> col = l&15, vgpr j holds B[2*kh + j][col]
//   == emb[c0+col][kbase + 2*kh + j]  (two consecutive floats -> one b64 load).
// C (16x16 f32): lane l -> col = l&15, vgpr j -> row = j + 8*(l>>4).
__global__ void __launch_bounds__(256)
vq_argmin(const float* __restrict__ x,        // NCHW [64,64,32,32]
          const float* __restrict__ emb,      // [1024,64]
          const float* __restrict__ halfnorm, // [1024]
          int* __restrict__ idx)              // [65536]
{
    const int lane  = threadIdx.x & 31;
    const int wave  = threadIdx.x >> 5;
    const int tile  = blockIdx.x * (blockDim.x >> 5) + wave; // 0..2047
    const int n0    = tile * 32;
    const int batch = n0 >> 10;      // n0 / 1024 (32-row tiles never cross a batch)
    const int hw0   = n0 & 1023;

    const int m  = lane & 15;
    const int kh = lane >> 4;        // 0 or 1

    // Load both A tiles once: rows n0+m and n0+16+m, d = 4*kk + 2*kh + {0,1}
    const float* xb0 = x + batch * (EMB_D * HW) + (hw0 + m);
    const float* xb1 = xb0 + 16;
    v2f a0[16], a1[16];
#pragma unroll
    for (int kk = 0; kk < 16; ++kk) {
        int d = 4 * kk + 2 * kh;
        a0[kk].x = xb0[d * HW];
        a0[kk].y = xb0[(d + 1) * HW];
        a1[kk].x = xb1[d * HW];
        a1[kk].y = xb1[(d + 1) * HW];
    }

    float best0[8], best1[8];
    int   bidx0[8], bidx1[8];
#pragma unroll
    for (int j = 0; j < 8; ++j) {
        best0[j] = -3.4e38f; bidx0[j] = 0;
        best1[j] = -3.4e38f; bidx1[j] = 0;
    }

    for (int t = 0; t < EMB_K / 16; ++t) {
        const int    col   = t * 16 + m;
        const float* ebase = emb + col * EMB_D + 2 * kh;
        const float  nbias = -halfnorm[col];

        v8f c0, c1;
#pragma unroll
        for (int j = 0; j < 8; ++j) { c0[j] = nbias; c1[j] = nbias; }

#pragma unroll
        for (int kk = 0; kk < 16; ++kk) {
            v2f bf = *(const v2f*)(ebase + 4 * kk);
            c0 = __builtin_amdgcn_wmma_f32_16x16x4_f32(
                    false, a0[kk], false, bf, (short)0, c0, false, false);
            c1 = __builtin_amdgcn_wmma_f32_16x16x4_f32(
                    false, a1[kk], false, bf, (short)0, c1, false, false);
        }

#pragma unroll
        for (int j = 0; j < 8; ++j) {
            if (c0[j] > best0[j]) { best0[j] = c0[j]; bidx0[j] = col; }
            if (c1[j] > best1[j]) { best1[j] = c1[j]; bidx1[j] = col; }
        }
    }

    // Per-row argmax across the 16 lanes sharing a row (butterfly, ties -> lowest idx).
#pragma unroll
    for (int j = 0; j < 8; ++j) {
        float v0 = best0[j]; int i0 = bidx0[j];
        float v1 = best1[j]; int i1 = bidx1[j];
#pragma unroll
        for (int s = 1; s <= 8; s <<= 1) {
            float vo0 = __shfl_xor(v0, s, 32);
            int   io0 = __shfl_xor(i0, s, 32);
            if (vo0 > v0 || (vo0 == v0 && io0 < i0)) { v0 = vo0; i0 = io0; }
            float vo1 = __shfl_xor(v1, s, 32);
            int   io1 = __shfl_xor(i1, s, 32);
            if (vo1 > v1 || (vo1 == v1 && io1 < i1)) { v1 = vo1; i1 = io1; }
        }
        if (m == 0) {
            const int row = j + 8 * kh;
            idx[n0 + row]      = i0;
            idx[n0 + 16 + row] = i1;
        }
    }
}

// ---------------------------------------------------------------------------
// Gather quantized output (NCHW), per-block loss partial, histogram.
__global__ void __launch_bounds__(256)
vq_gather(const float* __restrict__ x, const float* __restrict__ emb,
          const int* __restrict__ idx, float* __restrict__ out,
          float* __restrict__ partial, unsigned* __restrict__ counts)
{
    const int e  = blockIdx.x * blockDim.x + threadIdx.x; // 0..N_ELEMS-1
    const int hw = e & 1023;
    const int d  = (e >> 10) & 63;
    const int b  = e >> 16;
    const int n  = (b << 10) + hw;

    const int k  = idx[n];
    const float q  = emb[k * EMB_D + d];
    const float xv = x[e];
    out[e] = q;
    const float diff = q - xv;
    float s = diff * diff;

    // wave reduce, then block reduce -> deterministic per-block partial
#pragma unroll
    for (int off = 16; off > 0; off >>= 1) s += __shfl_down(s, off, 32);
    __shared__ float red[8];
    const int lane = threadIdx.x & 31;
    const int wave = threadIdx.x >> 5;
    if (lane == 0) red[wave] = s;
    __syncthreads();
    if (threadIdx.x == 0) {
        float tot = 0.f;
#pragma unroll
        for (int i = 0; i < 8; ++i) tot += red[i];
        partial[blockIdx.x] = tot;
    }

    if (d == 0) atomicAdd(&counts[k], 1u); // integer atomics: deterministic
}

// ---------------------------------------------------------------------------
// Single block: deterministic reductions for loss and perplexity.
__global__ void __launch_bounds__(1024)
vq_finalize(const unsigned* __restrict__ counts, const float* __restrict__ partial,
            float* __restrict__ tail)
{
    __shared__ float red[1024];
    const int t = threadIdx.x;

    float s = 0.f;
#pragma unroll
    for (int i = 0; i < 16; ++i) s += partial[t + i * 1024];
    red[t] = s;
    __syncthreads();
    for (int off = 512; off > 0; off >>= 1) {
        if (t < off) red[t] += red[t + off];
        __syncthreads();
    }
    const float loss_sum = red[0];
    __syncthreads();

    const float p = (float)counts[t] * (1.0f / (float)N_ROWS);
    red[t] = p * logf(p + 1e-10f);
    __syncthreads();
    for (int off = 512; off > 0; off >>= 1) {
        if (t < off) red[t] += red[t + off];
        __syncthreads();
    }

    if (t == 0) {
        tail[0] = 0.25f * loss_sum * (1.0f / (float)N_ELEMS); // commitment loss
        tail[1] = expf(-red[0]);                              // perplexity
    }
}

// ---------------------------------------------------------------------------
extern "C" void kernel_launch(void* const* d_in, const int* in_sizes, int n_in,
                              void* d_out, int out_size, void* d_ws, size_t ws_size,
                              hipStream_t stream) {
    const float* x   = (const float*)d_in[0]; // [64,64,32,32]
    const float* emb = (const float*)d_in[1]; // [1024,64]
    float* out  = (float*)d_out;              // quantized (NCHW) ++ [loss, perplexity]
    float* tail = out + N_ELEMS;

    char* ws = (char*)d_ws;
    unsigned* counts   = (unsigned*)(ws);           // 1024 * 4B
    float*    halfnorm = (float*)(ws + 4096);       // 1024 * 4B
    float*    partial  = (float*)(ws + 8192);       // 16384 * 4B
    int*      idx      = (int*)(ws + 8192 + 65536); // 65536 * 4B

    vq_init    <<<4,     256, 0, stream>>>(counts);
    vq_norms   <<<4,     256, 0, stream>>>(emb, halfnorm);
    vq_argmin  <<<256,   256, 0, stream>>>(x, emb, halfnorm, idx);   // 2048 waves * 2048 WMMA
    vq_gather  <<<16384, 256, 0, stream>>>(x, emb, idx, out, partial, counts);
    vq_finalize<<<1,    1024, 0, stream>>>(counts, partial, tail);
}